// GrayScott_Grad_2_33767032881350
// MI455X (gfx1250) — compile-verified
//
#include <hip/hip_runtime.h>
#include <stdint.h>

// Gray-Scott gradient: F_A = DA*lapA - S*A^2 + (MU+RHO)*A
//                      F_S = DS*lapS + S*A^2 - RHO*(1-S)
// plus verbatim copy of A_S as second output.
// Memory-bound (~1 FLOP/byte) -> CDNA5 async-DMA (ASYNCcnt) staging into LDS,
// float4 compute, non-temporal B128 stores (outputs are write-once; keep the
// 64 MiB input resident in the 192 MB L2 for halo re-reads).
// No WMMA by design (matrix reformulation inflates FLOPs 256x, ~10x slower
// than the ~8.7 us bandwidth floor at 23.3 TB/s).

#define W        256
#define TH       16                 // image rows per block
#define LR       (TH + 2)           // rows staged incl. halo
#define PLANE    (W * W)            // 65536 elements per channel image
#define CH_TILE  (LR * W)           // 4608 floats per channel LDS tile
#define NBATCH   128
#define F_ELEMS  (NBATCH * 2 * PLANE)   // 16,777,216

typedef float v4f __attribute__((ext_vector_type(4)));

__device__ __forceinline__ void async_load16(uint32_t lds_byte, uint32_t glb_byte,
                                             const float* base) {
    // GVS mode: 64-bit SGPR base + 32-bit per-lane byte offset -> 16B into LDS.
    // Tracked with ASYNCcnt, no VGPR data staging.
    asm volatile("global_load_async_to_lds_b128 %0, %1, %2"
                 :
                 : "v"(lds_byte), "v"(glb_byte), "s"(base)
                 : "memory");
}

__device__ __forceinline__ void store_nt16(float* p, v4f v) {
    __builtin_nontemporal_store(v, (v4f*)p);   // global_store_b128 ... th:TH_STORE_NT
}

__global__ __launch_bounds__(256)
void gray_scott_grad_kernel(const float* __restrict__ A_S, float* __restrict__ out) {
    __shared__ float lds[2 * CH_TILE];          // [A tile | S tile], 36 KiB

    const int tid  = threadIdx.x;
    const int b    = blockIdx.x >> 4;           // batch 0..127
    const int tile = blockIdx.x & 15;           // row tile 0..15
    const int r0   = tile * TH;

    const uint32_t lds_base = (uint32_t)(uintptr_t)(&lds[0]);
    const uint32_t plane_a  = (uint32_t)(b * 2 * PLANE);   // element offset of A plane

    // ---- Stage LR x 256 rows of both channels into LDS via async DMA ----
    // 2 ch * 18 rows * 64 chunks(16B) = 2304 chunk ops -> 9 per thread.
    #pragma unroll
    for (int i = 0; i < 9; ++i) {
        int idx = tid + i * 256;                // 0..2303
        int c   = (idx >= LR * 64) ? 1 : 0;     // LR*64 = 1152
        int k   = idx - c * (LR * 64);
        int lr  = k >> 6;                       // staged row 0..17
        int cx  = (k & 63) << 2;                // element x of 16B chunk
        int gr  = r0 - 1 + lr;                  // edge-pad clamp folded into DMA src
        gr = gr < 0 ? 0 : (gr > 255 ? 255 : gr);
        uint32_t lds_byte = lds_base + (uint32_t)((c * CH_TILE + lr * W + cx) << 2);
        uint32_t glb_byte = (plane_a + (uint32_t)(c * PLANE + gr * W + cx)) << 2;
        async_load16(lds_byte, glb_byte, A_S);
    }
    asm volatile("s_wait_asynccnt 0" ::: "memory");
    __syncthreads();

    // ---- Compute: each thread handles float4 pixels on 4 rows ----
    const int x4    = (tid & 63) << 2;          // 0..252
    const int rbase = tid >> 6;                 // 0..3

    const float* ldsA = &lds[0];
    const float* ldsS = &lds[CH_TILE];

    const float DA    = 0.00025f;
    const float DSc   = 0.0005f;
    const float MURHO = 0.065f + 0.04f;         // MU + RHO
    const float RHO   = 0.04f;
    const float INVH2 = 3844.0f;                // (N-1)^2, N=63

    float*       outF = out;                    // F, 16,777,216 floats
    float*       outC = out + F_ELEMS;          // copy of A_S

    #pragma unroll
    for (int rr = 0; rr < 4; ++rr) {
        const int rl   = rbase + rr * 4;        // local image row 0..15
        const int lrow = rl + 1;                // center row in LDS (halo offset)

        const v4f   cA = *(const v4f*)&ldsA[lrow * W + x4];
        const v4f   uA = *(const v4f*)&ldsA[(lrow - 1) * W + x4];
        const v4f   dA = *(const v4f*)&ldsA[(lrow + 1) * W + x4];
        const float lA = (x4 == 0)       ? cA.x : ldsA[lrow * W + x4 - 1];
        const float rA = (x4 == W - 4)   ? cA.w : ldsA[lrow * W + x4 + 4];

        const v4f   cS = *(const v4f*)&ldsS[lrow * W + x4];
        const v4f   uS = *(const v4f*)&ldsS[(lrow - 1) * W + x4];
        const v4f   dS = *(const v4f*)&ldsS[(lrow + 1) * W + x4];
        const float lS = (x4 == 0)       ? cS.x : ldsS[lrow * W + x4 - 1];
        const float rS = (x4 == W - 4)   ? cS.w : ldsS[lrow * W + x4 + 4];

        const float a6[6]  = { lA, cA.x, cA.y, cA.z, cA.w, rA };
        const float s6[6]  = { lS, cS.x, cS.y, cS.z, cS.w, rS };
        const float upA[4] = { uA.x, uA.y, uA.z, uA.w };
        const float dnA[4] = { dA.x, dA.y, dA.z, dA.w };
        const float upS[4] = { uS.x, uS.y, uS.z, uS.w };
        const float dnS[4] = { dS.x, dS.y, dS.z, dS.w };

        v4f fa, fs;
        #pragma unroll
        for (int i = 0; i < 4; ++i) {
            const float ac   = a6[i + 1];
            const float sc   = s6[i + 1];
            const float lapA = (4.0f * ac - upA[i] - dnA[i] - a6[i] - a6[i + 2]) * INVH2;
            const float lapS = (4.0f * sc - upS[i] - dnS[i] - s6[i] - s6[i + 2]) * INVH2;
            const float sa2  = sc * ac * ac;
            fa[i] = DA  * lapA - sa2 + MURHO * ac;
            fs[i] = DSc * lapS + sa2 - RHO * (1.0f - sc);
        }

        const int off = b * 2 * PLANE + (r0 + rl) * W + x4;   // A-plane element index
        store_nt16(&outF[off],         fa);
        store_nt16(&outF[off + PLANE], fs);
        store_nt16(&outC[off],         cA);                   // A_S copy, free
        store_nt16(&outC[off + PLANE], cS);
    }
}

extern "C" void kernel_launch(void* const* d_in, const int* in_sizes, int n_in,
                              void* d_out, int out_size, void* d_ws, size_t ws_size,
                              hipStream_t stream) {
    (void)in_sizes; (void)n_in; (void)out_size; (void)d_ws; (void)ws_size;
    const float* A_S = (const float*)d_in[0];
    float*       out = (float*)d_out;
    // 128 batches * 16 row-tiles = 2048 blocks, 256 threads (8 wave32s) each.
    gray_scott_grad_kernel<<<dim3(NBATCH * 16), dim3(256), 0, stream>>>(A_S, out);
}